// GNNEncoder_42279658062119
// MI455X (gfx1250) — compile-verified
//
#include <hip/hip_runtime.h>

#define NN 50000
#define NE 640000

typedef __attribute__((ext_vector_type(2))) float v2f;
typedef __attribute__((ext_vector_type(8))) float v8f;

__global__ void zero_f32(float* __restrict__ p, int n) {
    int i = blockIdx.x * blockDim.x + threadIdx.x;
    if (i < n) p[i] = 0.0f;
}

__global__ void count_deg(const int* __restrict__ dst, float* __restrict__ cnt, int E) {
    int e = blockIdx.x * blockDim.x + threadIdx.x;
    if (e < E) atomicAdd(cnt + dst[e], 1.0f);
}

__global__ void invert_cnt(float* __restrict__ cnt, int n) {
    int i = blockIdx.x * blockDim.x + threadIdx.x;
    if (i < n) cnt[i] = 1.0f / fmaxf(cnt[i], 1.0f);
}

// One thread handles one edge x 4 channels: float4 gather from x[src], 4 f32 atomics to agg[dst].
template <int D>
__global__ void scatter_add(const float* __restrict__ x, const int* __restrict__ src,
                            const int* __restrict__ dst, float* __restrict__ agg, int E) {
    const int VPE = D / 4;
    int t = blockIdx.x * blockDim.x + threadIdx.x;
    if (t >= E * VPE) return;
    int e  = t / VPE;
    int c4 = (t % VPE) * 4;
    int s = src[e];
    int d = dst[e];
    const float4 v = *reinterpret_cast<const float4*>(x + (size_t)s * D + c4);
    float* a = agg + (size_t)d * D + c4;
    atomicAdd(a + 0, v.x);
    atomicAdd(a + 1, v.y);
    atomicAdd(a + 2, v.z);
    atomicAdd(a + 3, v.w);
}

// out[n][m] = act( sum_k (agg[n][k]*inv_cnt[n]) * Wl[k][m] + sum_k xin[n][k] * Wr[k][m] + bias[m] )
// One wave32 computes a 16x16 tile via V_WMMA_F32_16X16X4_F32, accumulating over K in chunks of 4.
// A 16x4 f32 frag:  lane&15 = row M, half=lane>>4 selects K pair {0,1}/{2,3} -> v2f.
// B 4x16  f32 frag: lane&15 = col N, same half->K mapping -> v2f.
// C/D 16x16 f32:    VGPR i -> row i + 8*half, lane&15 = col.
template <int K, int M, int RELU>
__global__ __launch_bounds__(256) void sage_combine(
    const float* __restrict__ agg, const float* __restrict__ xin,
    const float* __restrict__ icnt,
    const float* __restrict__ wl, const float* __restrict__ wr,
    const float* __restrict__ bias, float* __restrict__ out) {
    const int ROW_TILES = NN / 16;
    const int COL_TILES = M / 16;
    int wave = (blockIdx.x * blockDim.x + threadIdx.x) >> 5;  // uniform per wave
    if (wave >= ROW_TILES * COL_TILES) return;
    int rowTile = wave % ROW_TILES;
    int colTile = wave / ROW_TILES;

    int lane = threadIdx.x & 31;
    int half = lane >> 4;
    int lr   = lane & 15;
    int row  = rowTile * 16 + lr;  // A-fragment row for this lane
    int col  = colTile * 16 + lr;  // B/D column for this lane
    float ic = icnt[row];

    v8f c = {0.f, 0.f, 0.f, 0.f, 0.f, 0.f, 0.f, 0.f};

    // mean @ Wl  (mean folded in by scaling A rows)
    const float* arow = agg + (size_t)row * K;
#pragma unroll
    for (int kb = 0; kb < K; kb += 4) {
        int k0 = kb + 2 * half;
        v2f a, b;
        a.x = arow[k0] * ic;
        a.y = arow[k0 + 1] * ic;
        b.x = wl[(size_t)k0 * M + col];
        b.y = wl[(size_t)(k0 + 1) * M + col];
        c = __builtin_amdgcn_wmma_f32_16x16x4_f32(false, a, false, b, (short)0, c, false, false);
    }
    // xin @ Wr
    const float* xrow = xin + (size_t)row * K;
#pragma unroll
    for (int kb = 0; kb < K; kb += 4) {
        int k0 = kb + 2 * half;
        v2f a, b;
        a.x = xrow[k0];
        a.y = xrow[k0 + 1];
        b.x = wr[(size_t)k0 * M + col];
        b.y = wr[(size_t)(k0 + 1) * M + col];
        c = __builtin_amdgcn_wmma_f32_16x16x4_f32(false, a, false, b, (short)0, c, false, false);
    }

    float bv = bias[col];
#pragma unroll
    for (int i = 0; i < 8; ++i) {
        float v = c[i] + bv;
        if (RELU) v = fmaxf(v, 0.0f);
        int orow = rowTile * 16 + i + 8 * half;
        out[(size_t)orow * M + col] = v;
    }
}

static inline int cdiv(long long a, int b) { return (int)((a + b - 1) / b); }

extern "C" void kernel_launch(void* const* d_in, const int* in_sizes, int n_in,
                              void* d_out, int out_size, void* d_ws, size_t ws_size,
                              hipStream_t stream) {
    (void)in_sizes; (void)n_in; (void)out_size; (void)ws_size;

    const float* x   = (const float*)d_in[0];
    const int*   ei  = (const int*)d_in[1];
    const int*   src = ei;
    const int*   dst = ei + NE;
    const float* w1l = (const float*)d_in[2];
    const float* b1  = (const float*)d_in[3];
    const float* w1r = (const float*)d_in[4];
    const float* w2l = (const float*)d_in[5];
    const float* b2  = (const float*)d_in[6];
    const float* w2r = (const float*)d_in[7];
    const float* w3l = (const float*)d_in[8];
    const float* b3  = (const float*)d_in[9];
    const float* w3r = (const float*)d_in[10];
    float* out = (float*)d_out;

    float* ws   = (float*)d_ws;
    float* cnt  = ws;                           // [NN]        (becomes inv_cnt)
    float* agg  = cnt + NN;                     // [NN * 128]  (reused per layer)
    float* h1   = agg + (size_t)NN * 128;       // [NN * 64]
    float* h2   = h1 + (size_t)NN * 64;         // [NN * 128]

    const int T = 256;

    // degree counts -> inverse counts (float, matching reference segment_sum of ones)
    zero_f32<<<cdiv(NN, T), T, 0, stream>>>(cnt, NN);
    count_deg<<<cdiv(NE, T), T, 0, stream>>>(dst, cnt, NE);
    invert_cnt<<<cdiv(NN, T), T, 0, stream>>>(cnt, NN);

    // ---- Layer 1: 32 -> 64, ReLU ----
    zero_f32<<<cdiv((long long)NN * 32, T), T, 0, stream>>>(agg, NN * 32);
    scatter_add<32><<<cdiv((long long)NE * 8, T), T, 0, stream>>>(x, src, dst, agg, NE);
    {
        long long waves = (long long)(NN / 16) * (64 / 16);
        sage_combine<32, 64, 1><<<cdiv(waves * 32, T), T, 0, stream>>>(agg, x, cnt, w1l, w1r, b1, h1);
    }

    // ---- Layer 2: 64 -> 128, ReLU ----
    zero_f32<<<cdiv((long long)NN * 64, T), T, 0, stream>>>(agg, NN * 64);
    scatter_add<64><<<cdiv((long long)NE * 16, T), T, 0, stream>>>(h1, src, dst, agg, NE);
    {
        long long waves = (long long)(NN / 16) * (128 / 16);
        sage_combine<64, 128, 1><<<cdiv(waves * 32, T), T, 0, stream>>>(agg, h1, cnt, w2l, w2r, b2, h2);
    }

    // ---- Layer 3: 128 -> 64, no activation ----
    zero_f32<<<cdiv((long long)NN * 128, T), T, 0, stream>>>(agg, NN * 128);
    scatter_add<128><<<cdiv((long long)NE * 32, T), T, 0, stream>>>(h2, src, dst, agg, NE);
    {
        long long waves = (long long)(NN / 16) * (64 / 16);
        sage_combine<128, 64, 0><<<cdiv(waves * 32, T), T, 0, stream>>>(agg, h2, cnt, w3l, w3r, b3, out);
    }
}